// Encoder_48593259987405
// MI455X (gfx1250) — compile-verified
//
#include <hip/hip_runtime.h>

typedef float v2f __attribute__((ext_vector_type(2)));
typedef float v4f __attribute__((ext_vector_type(4)));
typedef float v8f __attribute__((ext_vector_type(8)));

#define B_   8
#define L_   2048
#define D_   512
#define TMAX 200.0f
#define LOG2E 1.4426950408889634f

__device__ __forceinline__ float fexp2(float x) {
#if __has_builtin(__builtin_amdgcn_exp2f)
    return __builtin_amdgcn_exp2f(x);      // raw v_exp_f32
#else
    return exp2f(x);
#endif
}
__device__ __forceinline__ float frcp(float x) {
#if __has_builtin(__builtin_amdgcn_rcpf)
    return __builtin_amdgcn_rcpf(x);       // raw v_rcp_f32
#else
    return 1.0f / x;
#endif
}

// ---------------------------------------------------------------------------
// Prologue: fold softplus/sigmoid params into the 3 exp2/rcp coefficients,
// once for the whole grid (they are wave-uniform constants otherwise
// recomputed with IEEE-division chains by every wave).
//   score = exp2(d^2*c1 + 1) * rcp(1 + exp2(|d|*c2 + c3))
// ---------------------------------------------------------------------------
__global__ void tpp_coeffs(const float* __restrict__ lsp,
                           const float* __restrict__ gp,
                           float* __restrict__ coeffs) {
    const float ls    = __logf(1.0f + __expf(lsp[0]));    // softplus
    const float gl    = 1.0f / (1.0f + __expf(-gp[0]));   // sigmoid
    const float inv_s = 1.0f + __expf(-gp[1]);            // 1/sigmoid
    coeffs[0] = -LOG2E / (ls * ls);                       // c1 (on d^2)
    coeffs[1] = -2.0f * LOG2E * inv_s * (1.0f / TMAX);    // c2 (on |d|)
    coeffs[2] =  2.0f * LOG2E * inv_s * gl;               // c3 (bias)
}

// ---------------------------------------------------------------------------
// scores + t_diff. One wave handles a 16x64 strip: 4 adjacent 16x16 tiles.
// Per tile, one V_WMMA_F32_16X16X4_F32 computes the exact pairwise-difference
// tile as the rank-2 product  t·1^T + 1·(-t)^T  (A col0=t_rows, col1=1;
// B row0=1, row1=-t_cols; K=2,3 zeroed on lanes 16-31).
// Tiles strictly above the diagonal skip all score math (scores == 0).
// All outputs written with non-temporal stores (302 MB > 192 MB L2,
// never re-read).
// ---------------------------------------------------------------------------
__global__ __launch_bounds__(256) void tpp_scores_tdiff(
    const float* __restrict__ t,       // [B, L] event times
    const float* __restrict__ coeffs,  // [3] from tpp_coeffs
    float* __restrict__ scores,        // [B, L, L]
    float* __restrict__ tdiff)         // [B, L, L]
{
    const float c1 = coeffs[0];
    const float c2 = coeffs[1];
    const float c3 = coeffs[2];

    const int wave = blockIdx.x * 8 + (threadIdx.x >> 5);
    const int lane = threadIdx.x & 31;
    const int b    = wave >> 12;          // 128*32 strips per batch
    const int rem  = wave & 4095;
    const int ti   = rem >> 5;            // row tile 0..127
    const int tj0  = (rem & 31) << 2;     // first of 4 col tiles

    const int  ln = lane & 15;
    const bool lo = lane < 16;

    const float trow = lo ? t[b * L_ + ti * 16 + ln] : 0.0f;
    const v2f a = { trow, lo ? 1.0f : 0.0f };   // A reused for all 4 WMMAs

    const int i0 = ti * 16 + ((lane >> 4) << 3);      // first row this half-wave
    const int j0 = tj0 * 16 + ln;                     // first col this lane
    const size_t base = ((size_t)b * L_ + i0) * L_ + j0;
    float* tdp = tdiff  + base;
    float* scp = scores + base;

#pragma unroll
    for (int tt = 0; tt < 4; ++tt) {
        const int tjTile = tj0 + tt;
        const float tcol = lo ? t[b * L_ + tjTile * 16 + ln] : 0.0f;
        const v2f bm = { lo ? 1.0f : 0.0f, -tcol };
        const v8f cz = {0.f,0.f,0.f,0.f,0.f,0.f,0.f,0.f};
        const v8f d = __builtin_amdgcn_wmma_f32_16x16x4_f32(
                          false, a, false, bm, (short)0, cz, false, false);

        if (tjTile > ti) {
            // strictly above diagonal: t_diff still needed, scores all zero
#pragma unroll
            for (int r = 0; r < 8; ++r) {
                __builtin_nontemporal_store(d[r], tdp + tt * 16 + (size_t)r * L_);
                __builtin_nontemporal_store(0.0f, scp + tt * 16 + (size_t)r * L_);
            }
        } else {
            const bool diag = (tjTile == ti);
            const int  j    = j0 + tt * 16;
#pragma unroll
            for (int r = 0; r < 8; ++r) {
                const float diff = d[r];
                __builtin_nontemporal_store(diff, tdp + tt * 16 + (size_t)r * L_);

                const float ad = fabsf(diff);
                const float e1 = fexp2(fmaf(ad * ad, c1, 1.0f));  // 2*kern
                const float e2 = fexp2(fmaf(ad, c2, c3));         // exp(-2z)
                float sc = e1 * frcp(1.0f + e2);                  // kern*gate
                if (diag && (j > i0 + r)) sc = 0.0f;              // tril mask
                __builtin_nontemporal_store(sc, scp + tt * 16 + (size_t)r * L_);
            }
        }
    }
}

// ---------------------------------------------------------------------------
// Temporal positional encoding. One float4 NT store per thread.
//   pos(d) = 10000^(d/256)  ->  te = t * exp2(-d * log2(10000)/256)
// ---------------------------------------------------------------------------
__global__ __launch_bounds__(256) void tpp_embedding(
    const float* __restrict__ t,      // [B, L]
    float* __restrict__ emb)          // [B, L, D]
{
    const int idx = blockIdx.x * 256 + threadIdx.x;   // B*L*D/4 threads
    const int bl  = idx >> 7;                          // D/4 = 128
    const int d0  = (idx & 127) * 4;

    const float tv = t[bl];

    v4f out;
#pragma unroll
    for (int k = 0; k < 4; ++k) {
        const int   dd  = d0 + k;
        const float arg = tv * fexp2((float)dd * -0.05190512648261504f);
        out[k] = (dd & 1) ? __cosf(arg) : __sinf(arg);
    }
    __builtin_nontemporal_store(out, (v4f*)(emb + (size_t)idx * 4));
}

// ---------------------------------------------------------------------------
extern "C" void kernel_launch(void* const* d_in, const int* in_sizes, int n_in,
                              void* d_out, int out_size, void* d_ws, size_t ws_size,
                              hipStream_t stream) {
    // inputs: [0] event_type (unused), [1] event_time f32, [2] ls param, [3] gate params
    const float* t   = (const float*)d_in[1];
    const float* lsp = (const float*)d_in[2];
    const float* gp  = (const float*)d_in[3];

    // outputs concatenated: scores [B,L,L] | embedding [B,L,D] | t_diff [B,L,L]
    float* scores = (float*)d_out;
    float* emb    = scores + (size_t)B_ * L_ * L_;
    float* tdiff  = emb    + (size_t)B_ * L_ * D_;

    float* coeffs = (float*)d_ws;   // 3 floats, rewritten every call

    const int embBlocks  = (B_ * L_ * D_ / 4) / 256;               // 8192
    const int tileBlocks = (B_ * (L_ / 16) * (L_ / 16) / 4) / 8;   // 4096

    tpp_coeffs      <<<1,          1,   0, stream>>>(lsp, gp, coeffs);
    tpp_embedding   <<<embBlocks,  256, 0, stream>>>(t, emb);
    tpp_scores_tdiff<<<tileBlocks, 256, 0, stream>>>(t, coeffs, scores, tdiff);
}